// GATLayer_72782515798576
// MI455X (gfx1250) — compile-verified
//
#include <hip/hip_runtime.h>
#include <math.h>

typedef float v2f __attribute__((ext_vector_type(2)));
typedef float v4f __attribute__((ext_vector_type(4)));
typedef float v8f __attribute__((ext_vector_type(8)));

#define N_NODES 8192
#define F_IN    256
#define F_OUT   128
#define NEG_SLOPE 0.2f
#define CHUNK   128
#define APITCH  132   // 128 + 4 floats padding -> conflict-free ds_load_b64 A-frags

__device__ __forceinline__ float lrelu(float u) {
    return u > 0.0f ? u : NEG_SLOPE * u;
}

// ---------------------------------------------------------------------------
// Kernel 1: h = x @ W^T  via V_WMMA_F32_16X16X4_F32
// grid.x = 512 i-tiles of 16 rows; block = 256 (8 waves; wave w -> n-tile w)
// ---------------------------------------------------------------------------
__global__ __launch_bounds__(256) void k_h_gemm(const float* __restrict__ x,
                                                const float* __restrict__ W,
                                                float* __restrict__ h) {
    const int tid  = threadIdx.x;
    const int wave = tid >> 5;
    const int lane = tid & 31;
    const int i0   = blockIdx.x * 16;
    const int n0   = wave * 16;
    const int row  = lane & 15;          // M for A-frag, N for B-frag
    const int kk   = (lane >> 4) << 1;   // lanes 16..31 hold K+2,K+3

    const float* xrow = x + (size_t)(i0 + row) * F_IN;  // A: x rows
    const float* wrow = W + (size_t)(n0 + row) * F_IN;  // B = W^T: W[n][k],k+1 contiguous

    v8f acc = {};
    for (int k0 = 0; k0 < F_IN; k0 += 4) {
        v2f a = *(const v2f*)(xrow + k0 + kk);
        v2f b = *(const v2f*)(wrow + k0 + kk);
        acc = __builtin_amdgcn_wmma_f32_16x16x4_f32(false, a, false, b,
                                                    (short)0, acc, false, false);
    }
    // D layout: VGPR v -> M = v + 8*(lane>=16), N = lane&15
    const int mhi = (lane >> 4) * 8;
    #pragma unroll
    for (int v = 0; v < 8; ++v)
        h[(size_t)(i0 + mhi + v) * F_OUT + n0 + row] = acc[v];
}

// ---------------------------------------------------------------------------
// Kernel 2: s_src = h@a[:F], s_dst = h@a[F:]
// ---------------------------------------------------------------------------
__global__ __launch_bounds__(256) void k_scores(const float* __restrict__ h,
                                                const float* __restrict__ a,
                                                float* __restrict__ sSrc,
                                                float* __restrict__ sDst) {
    const int i = blockIdx.x * blockDim.x + threadIdx.x;
    if (i >= N_NODES) return;
    const float* hr = h + (size_t)i * F_OUT;
    float ss = 0.f, sd = 0.f;
    for (int c = 0; c < F_OUT; ++c) {
        const float hv = hr[c];
        ss = fmaf(hv, a[c], ss);
        sd = fmaf(hv, a[F_OUT + c], sd);
    }
    sSrc[i] = ss;
    sDst[i] = sd;
}

// ---------------------------------------------------------------------------
// Kernel 3: dmax = max_j s_dst[j]   (lrelu monotone => row max of e is
//           lrelu(s_src_i + dmax), exact softmax max-shift)
// ---------------------------------------------------------------------------
__global__ __launch_bounds__(1024) void k_dmax(const float* __restrict__ sDst,
                                               float* __restrict__ dmax) {
    __shared__ float red[1024];
    const int t = threadIdx.x;
    float m = -INFINITY;
    for (int j = t; j < N_NODES; j += 1024) m = fmaxf(m, sDst[j]);
    red[t] = m;
    __syncthreads();
    for (int s = 512; s > 0; s >>= 1) {
        if (t < s) red[t] = fmaxf(red[t], red[t + s]);
        __syncthreads();
    }
    if (t == 0) dmax[0] = red[0];
}

// ---------------------------------------------------------------------------
// Kernel 4: fused  alpha generation -> HBM (NT stores)  +  out = alpha @ h
// grid.x = 512 i-tiles; block = 256 (8 waves, wave w owns N-tile 16w)
// ---------------------------------------------------------------------------
__global__ __launch_bounds__(256) void k_attn(const float* __restrict__ h,
                                              const float* __restrict__ sSrc,
                                              const float* __restrict__ sDst,
                                              const float* __restrict__ dmaxp,
                                              float* __restrict__ outO,
                                              float* __restrict__ alphaO) {
    __shared__ float lsA[16 * APITCH];
    __shared__ float lsS[16], lsM[16], lsZ[16];
    __shared__ float lred[256];

    const int tid = threadIdx.x;
    const int i0  = blockIdx.x * 16;
    const float dmax = dmaxp[0];

    if (tid < 16) {
        const float s = sSrc[i0 + tid];
        lsS[tid] = s;
        lsM[tid] = lrelu(s + dmax);   // exact row max of e
    }
    __syncthreads();

    // ---- Phase 1: softmax normalizers Z_i (16 rows x 16 lanes each) ----
    {
        const int r = tid >> 4;
        const int l = tid & 15;
        const float s = lsS[r], m = lsM[r];
        float z = 0.f;
        for (int j = l; j < N_NODES; j += 16)
            z += __expf(lrelu(s + sDst[j]) - m);
        lred[tid] = z;
        __syncthreads();
        if (tid < 16) {
            float Z = 0.f;
            #pragma unroll
            for (int q = 0; q < 16; ++q) Z += lred[tid * 16 + q];
            lsZ[tid] = 1.0f / Z;
        }
        __syncthreads();
    }

    // ---- Phase 2: per 16x128 chunk: alpha -> LDS + HBM, then WMMA ----
    const int wave = tid >> 5;
    const int lane = tid & 31;
    const int n0   = wave * 16;
    const int bn   = lane & 15;          // N within tile (B/C/D frags)
    const int kk   = (lane >> 4) << 1;   // K sub-offset for A/B frags
    const int arow = lane & 15;          // M within tile (A frag)

    v8f acc = {};

    for (int j0 = 0; j0 < N_NODES; j0 += CHUNK) {
        // generate normalized alpha chunk: 512 v4f, 2 per thread
        #pragma unroll
        for (int rep = 0; rep < 2; ++rep) {
            const int f = tid + rep * 256;
            const int r = f >> 5;
            const int c = (f & 31) << 2;
            const float s = lsS[r], m = lsM[r], zi = lsZ[r];
            const v4f d = *(const v4f*)(sDst + j0 + c);
            v4f p;
            p.x = __expf(lrelu(s + d.x) - m) * zi;
            p.y = __expf(lrelu(s + d.y) - m) * zi;
            p.z = __expf(lrelu(s + d.z) - m) * zi;
            p.w = __expf(lrelu(s + d.w) - m) * zi;
            *(v4f*)(lsA + r * APITCH + c) = p;
            // 268 MB stream > 192 MB L2: write non-temporal
            __builtin_nontemporal_store(p,
                (v4f*)(alphaO + (size_t)(i0 + r) * N_NODES + j0 + c));
        }
        __syncthreads();

        // out-tile accumulation: A = alpha chunk (LDS), B = h rows (global/L2)
        const float* hb = h + (size_t)j0 * F_OUT + n0 + bn;
        for (int kc = 0; kc < CHUNK; kc += 4) {
            v2f a = *(const v2f*)(lsA + arow * APITCH + kc + kk);
            v2f b;
            b.x = hb[(size_t)(kc + kk) * F_OUT];
            b.y = hb[(size_t)(kc + kk + 1) * F_OUT];
            acc = __builtin_amdgcn_wmma_f32_16x16x4_f32(false, a, false, b,
                                                        (short)0, acc, false, false);
        }
        __syncthreads();
    }

    const int mhi = (lane >> 4) * 8;
    #pragma unroll
    for (int v = 0; v < 8; ++v)
        outO[(size_t)(i0 + mhi + v) * F_OUT + n0 + bn] = acc[v];
}

// ---------------------------------------------------------------------------
extern "C" void kernel_launch(void* const* d_in, const int* in_sizes, int n_in,
                              void* d_out, int out_size, void* d_ws, size_t ws_size,
                              hipStream_t stream) {
    const float* x = (const float*)d_in[0];   // (8192, 256)
    const float* W = (const float*)d_in[1];   // (128, 256)
    const float* a = (const float*)d_in[2];   // (256,)

    float* out   = (float*)d_out;                       // (8192, 128)
    float* alpha = (float*)d_out + (size_t)N_NODES * F_OUT; // (8192, 8192)

    char* ws = (char*)d_ws;
    float* h    = (float*)ws;                                   // 4 MB
    float* sSrc = (float*)(ws + (size_t)N_NODES * F_OUT * 4);   // 32 KB
    float* sDst = sSrc + N_NODES;                               // 32 KB
    float* dmax = sDst + N_NODES;                               // 4 B

    k_h_gemm<<<N_NODES / 16, 256, 0, stream>>>(x, W, h);
    k_scores<<<N_NODES / 256, 256, 0, stream>>>(h, a, sSrc, sDst);
    k_dmax<<<1, 1024, 0, stream>>>(sDst, dmax);
    k_attn<<<N_NODES / 16, 256, 0, stream>>>(h, sSrc, sDst, dmax, out, alpha);
}